// MultiLoRALayerMaskingHomEfficient_44933947850967
// MI455X (gfx1250) — compile-verified
//
#include <hip/hip_runtime.h>
#include <hip/hip_bf16.h>

// ---------------------------------------------------------------------------
// Fused multi-LoRA: out[b,n,o] = sum_r ( sum_i x[b,n,i]*B[id[b],r,i] ) * A[id[b],o,r]
// batch=8, n_tok=2048, d_in=d_out=4096, rank=16, SCALING = 16/16 = 1.0.
//
// Roofline: ~516 MB HBM traffic @ 23.3 TB/s ~ 22us; 4.3 GFLOP -> needs
// ~195 TFLOPS f32 to stay bandwidth-bound -> V_WMMA_F32_16X16X4_F32 for both
// rank-16 GEMM stages (full f32 precision). x/B tiles staged to LDS by the
// Tensor Data Mover (double-buffered, TENSORcnt-tracked) so DMA overlaps WMMA.
// ---------------------------------------------------------------------------

typedef __attribute__((ext_vector_type(2))) float v2f;
typedef __attribute__((ext_vector_type(4))) float v4f;
typedef __attribute__((ext_vector_type(8))) float v8f;

#define N_BATCH 8
#define N_TOK   2048
#define D_IN    4096
#define D_OUT   4096
#define RANK    16
#define SCALING 1.0f

#define TM   64     // tokens per workgroup (4 WMMA M-tiles)
#define KC   128    // d_in chunk staged in LDS per iteration
#define XSTR 132    // padded LDS row stride (TDM pad: 4 DW after every 128 DW)
#define NIT  (D_IN / KC)

#if __has_builtin(__builtin_amdgcn_tensor_load_to_lds)
#define USE_TDM 1
typedef unsigned int v4u __attribute__((ext_vector_type(4)));
typedef int          v8s __attribute__((ext_vector_type(8)));
typedef int          v4s __attribute__((ext_vector_type(4)));

// Low 32 bits of a generic pointer into LDS == LDS byte offset (flat aperture
// mapping keeps the low 32 address bits).
__device__ __forceinline__ unsigned lds_off(const void* p) {
    return (unsigned)(uintptr_t)p;
}

// 2D tile load, row-major f32: tile_rows x 128 elements, tensor line = 4096
// elements, LDS padding 4 DWORDs per 128 DWORDs -> effective row stride 132.
// D# layout per CDNA5 ISA 8.3/8.4 (group0: ctrl/lds/global/type, group1: dims).
// This toolchain exposes the 6-arg builtin: (g0, g1, g2, g3, g_ext, cpol);
// groups 2/3 (+ext) are unused for a 2D tile -> zero-filled.
__device__ __forceinline__ void tdm_load_2d(const float* g, unsigned lds,
                                            int tile_rows, unsigned tensor_rows) {
    unsigned long long ga = (unsigned long long)(uintptr_t)g;
    v4u g0;
    g0.x = 1u;                                        // count=1 user descriptor
    g0.y = lds;                                       // lds_addr (bytes)
    g0.z = (unsigned)(ga & 0xffffffffu);              // global_addr[31:0]
    g0.w = (unsigned)((ga >> 32) & 0x01ffffffu)       // global_addr[56:32]
         | (2u << 30);                                // type=2 ("image")
    v8s g1;
    g1[0] = (2 << 16)                                 // data_size = 4 bytes
          | (1 << 20)                                 // pad_enable
          | (6 << 22)                                 // pad_interval = 128 DW
          | (3 << 25);                                // pad_amount  = 4 DW
    g1[1] = (int)((D_IN & 0xffffu) << 16);            // tensor_dim0 = 4096
    g1[2] = (int)((tensor_rows & 0xffffu) << 16);     // tensor_dim1 lo16
    g1[3] = (int)((tensor_rows >> 16) & 0xffffu)      // tensor_dim1 hi16
          | (KC << 16);                               // tile_dim0 = 128
    g1[4] = tile_rows;                                // tile_dim1 (tile_dim2=0)
    g1[5] = D_IN;                                     // tensor_dim0_stride = 4096
    g1[6] = 0;
    g1[7] = 0;
    v4s z4 = {0, 0, 0, 0};
    v8s z8 = {0, 0, 0, 0, 0, 0, 0, 0};
    __builtin_amdgcn_tensor_load_to_lds(g0, g1, z4, z4, z8, 0);
}
#else
#define USE_TDM 0
#endif

__global__ __launch_bounds__(256)
void lora_fused_wmma(const float* __restrict__ x,
                     const float* __restrict__ A,
                     const float* __restrict__ B,
                     const int*   __restrict__ ids,
                     float*       __restrict__ out) {
    __shared__ float sX[2][TM * XSTR];       // double-buffered x chunk
    __shared__ float sB[2][RANK * XSTR];     // double-buffered B chunk
    __shared__ float sPart[8 * 32 * 9];      // stage-1 K-split partials
    __shared__ float sBx[4 * 16 * 18];       // Bx tile [mt][m][r], stride 18

    const int t    = threadIdx.x;
    const int w    = t >> 5;                 // wave id 0..7
    const int lane = t & 31;
    const int m16  = lane & 15;              // ISA layout: lane%16 = M(A)/N(B,D)
    const int hi   = lane >> 4;              // lane-half -> K pair / +8 rows
    const int dk   = hi << 1;

    const int b   = blockIdx.y;
    const int n0  = blockIdx.x * TM;
    const int aid = ids[b];

    const int mt = w >> 1;                   // token sub-tile 0..3
    const int kr = w & 1;                    // K split within staged chunk

    const float* xb = x + ((size_t)b * N_TOK + n0) * D_IN;
    const float* Bb = B + (size_t)aid * RANK * D_IN;

    // ---------------- Stage 1: Bx[m,r] = sum_i x[m,i] * B[r,i] ----------------
#if USE_TDM
    if (w == 0) {
        tdm_load_2d(xb, lds_off(&sX[0][0]), TM, 1u << 20);
        tdm_load_2d(Bb, lds_off(&sB[0][0]), RANK, RANK);
        __builtin_amdgcn_s_wait_tensorcnt(0);
    }
    __syncthreads();
#endif

    v8f acc0 = {}, acc1 = {};                // dual accumulators: 2x WMMA ILP
    for (int it = 0; it < NIT; ++it) {
        const int cur = it & 1;
#if USE_TDM
        // Kick DMA for chunk it+1 while all 8 waves compute chunk it.
        if (w == 0 && it + 1 < NIT) {
            tdm_load_2d(xb + (it + 1) * KC, lds_off(&sX[cur ^ 1][0]), TM, 1u << 20);
            tdm_load_2d(Bb + (it + 1) * KC, lds_off(&sB[cur ^ 1][0]), RANK, RANK);
        }
#else
        {   // manual staging fallback: coalesced float4 global->LDS
            const int kb = it * KC;
            #pragma unroll
            for (int j = 0; j < 8; ++j) {
                int f = t + 256 * j, row = f >> 5, c4 = f & 31;
                v4f v = *(const v4f*)(xb + (size_t)row * D_IN + kb + c4 * 4);
                *(v4f*)&sX[cur][row * XSTR + c4 * 4] = v;
            }
            #pragma unroll
            for (int j = 0; j < 2; ++j) {
                int f = t + 256 * j, row = f >> 5, c4 = f & 31;
                v4f v = *(const v4f*)(Bb + (size_t)row * D_IN + kb + c4 * 4);
                *(v4f*)&sB[cur][row * XSTR + c4 * 4] = v;
            }
        }
        __syncthreads();
#endif
        // 16 WMMAs/wave/chunk: A = x[16tok x 4k], B = B^T[4k x 16r].
        // Operand layout (ISA 7.12.2): lane%16 = row, VGPR pair K=2*(lane/16)+{0,1}.
        const float* xp = &sX[cur][(mt * 16 + m16) * XSTR + kr * 64 + dk];
        const float* bp = &sB[cur][m16 * XSTR + kr * 64 + dk];
        #pragma unroll
        for (int j = 0; j < 8; ++j) {
            v2f a0 = *(const v2f*)(xp + 8 * j);
            v2f b0 = *(const v2f*)(bp + 8 * j);
            v2f a1 = *(const v2f*)(xp + 8 * j + 4);
            v2f b1 = *(const v2f*)(bp + 8 * j + 4);
            acc0 = __builtin_amdgcn_wmma_f32_16x16x4_f32(
                false, a0, false, b0, (short)0, acc0, false, false);
            acc1 = __builtin_amdgcn_wmma_f32_16x16x4_f32(
                false, a1, false, b1, (short)0, acc1, false, false);
        }
#if USE_TDM
        if (w == 0) __builtin_amdgcn_s_wait_tensorcnt(0);
#endif
        __syncthreads();
    }
    v8f acc = acc0 + acc1;

    // Reduce the two K-split partials, transpose C-layout into sBx[m][r].
    #pragma unroll
    for (int v = 0; v < 8; ++v) sPart[(w * 32 + lane) * 9 + v] = acc[v];
    __syncthreads();
    if (kr == 0) {
        #pragma unroll
        for (int v = 0; v < 8; ++v) {
            float r = sPart[(w * 32 + lane) * 9 + v]
                    + sPart[((w + 1) * 32 + lane) * 9 + v];
            sBx[(mt * 16 + v + 8 * hi) * 18 + m16] = r;
        }
    }
    __syncthreads();

    // ---------------- Stage 2: out[m,o] = sum_r Bx[m,r] * A[o,r] ----------------
    v2f aq[4];                               // Bx as WMMA A-operand, loaded once
    #pragma unroll
    for (int q = 0; q < 4; ++q)
        aq[q] = *(const v2f*)&sBx[(mt * 16 + m16) * 18 + 4 * q + dk];

    const int orr = w & 1;                   // o-range split: 2 x 2048
    const float* Ab = A + (size_t)aid * D_OUT * RANK;
    float* ob = out + ((size_t)b * N_TOK + n0 + mt * 16) * D_OUT;

    for (int ot2 = 0; ot2 < 2048 / 32; ++ot2) {   // two o-tiles per iteration
        const int oA = orr * 2048 + ot2 * 32;
        const int oB = oA + 16;
        // Each lane owns one contiguous 16-float A row (4x float4); lane-half
        // selects .xy/.zw as the per-chunk B operand (K pair).
        const float* ra = Ab + (size_t)(oA + m16) * RANK;
        const float* rb = Ab + (size_t)(oB + m16) * RANK;
        v4f a0 = *(const v4f*)(ra + 0),  a1 = *(const v4f*)(ra + 4);
        v4f a2 = *(const v4f*)(ra + 8),  a3 = *(const v4f*)(ra + 12);
        v4f c0 = *(const v4f*)(rb + 0),  c1 = *(const v4f*)(rb + 4);
        v4f c2 = *(const v4f*)(rb + 8),  c3 = *(const v4f*)(rb + 12);

        v8f d0 = {}, d1 = {};                // two independent WMMA chains
        d0 = __builtin_amdgcn_wmma_f32_16x16x4_f32(false, aq[0], false, hi ? a0.zw : a0.xy, (short)0, d0, false, false);
        d1 = __builtin_amdgcn_wmma_f32_16x16x4_f32(false, aq[0], false, hi ? c0.zw : c0.xy, (short)0, d1, false, false);
        d0 = __builtin_amdgcn_wmma_f32_16x16x4_f32(false, aq[1], false, hi ? a1.zw : a1.xy, (short)0, d0, false, false);
        d1 = __builtin_amdgcn_wmma_f32_16x16x4_f32(false, aq[1], false, hi ? c1.zw : c1.xy, (short)0, d1, false, false);
        d0 = __builtin_amdgcn_wmma_f32_16x16x4_f32(false, aq[2], false, hi ? a2.zw : a2.xy, (short)0, d0, false, false);
        d1 = __builtin_amdgcn_wmma_f32_16x16x4_f32(false, aq[2], false, hi ? c2.zw : c2.xy, (short)0, d1, false, false);
        d0 = __builtin_amdgcn_wmma_f32_16x16x4_f32(false, aq[3], false, hi ? a3.zw : a3.xy, (short)0, d0, false, false);
        d1 = __builtin_amdgcn_wmma_f32_16x16x4_f32(false, aq[3], false, hi ? c3.zw : c3.xy, (short)0, d1, false, false);

        // D layout: lane%16 = column o, rows m = v + 8*(lane/16). Coalesced stores.
        #pragma unroll
        for (int v = 0; v < 8; ++v) {
            float* orow = ob + (size_t)(v + 8 * hi) * D_OUT + m16;
            orow[oA] = d0[v] * SCALING;
            orow[oB] = d1[v] * SCALING;
        }
    }
}

extern "C" void kernel_launch(void* const* d_in, const int* in_sizes, int n_in,
                              void* d_out, int out_size, void* d_ws, size_t ws_size,
                              hipStream_t stream) {
    const float* x   = (const float*)d_in[0];
    const float* A   = (const float*)d_in[1];
    const float* B   = (const float*)d_in[2];
    const int*   ids = (const int*)d_in[3];
    float* out = (float*)d_out;
    (void)in_sizes; (void)n_in; (void)out_size; (void)d_ws; (void)ws_size;

    dim3 grid(N_TOK / TM, N_BATCH, 1);   // 32 x 8 = 256 workgroups
    dim3 block(256, 1, 1);               // 8 wave32s
    lora_fused_wmma<<<grid, block, 0, stream>>>(x, A, B, ids, out);
}